// UOTPooling2_65386582114530
// MI455X (gfx1250) — compile-verified
//
#include <hip/hip_runtime.h>

#ifndef __has_builtin
#define __has_builtin(x) 0
#endif

#define N_NODES   16384
#define DFEAT     128
#define BSEG      256
#define KITER     40
#define CMAX      160      // max rows per segment (true max ~90; huge safety margin)
#define NTHREADS  256      // 8 wave32 waves
#define NWAVES    8
#define EPS_Y     1e-8f

typedef float        v2f __attribute__((ext_vector_type(2)));
typedef float        v8f __attribute__((ext_vector_type(8)));
typedef unsigned int v4u __attribute__((ext_vector_type(4)));
typedef int          v4i __attribute__((ext_vector_type(4)));
typedef int          v8i __attribute__((ext_vector_type(8)));

__device__ __forceinline__ float softplus_f(float v) {
  return (v > 20.0f) ? v : log1pf(expf(v));
}

__device__ __forceinline__ float wave_reduce_max(float v) {
#pragma unroll
  for (int m = 16; m >= 1; m >>= 1) v = fmaxf(v, __shfl_xor(v, m, 32));
  return v;
}
__device__ __forceinline__ float wave_reduce_sum(float v) {
#pragma unroll
  for (int m = 16; m >= 1; m >>= 1) v += __shfl_xor(v, m, 32);
  return v;
}

// One workgroup per segment. Lane = column (4 chunks of 32), wave strides rows.
// Entire 40-iteration scan runs out of LDS: no global sync, no global state traffic.
__global__ __launch_bounds__(NTHREADS)
void uot_pool_kernel(const float* __restrict__ gx,
                     const float* __restrict__ grho,
                     const float* __restrict__ ga1,
                     const float* __restrict__ ga2,
                     const float* __restrict__ ga3,
                     const int*   __restrict__ gbatch,
                     float* __restrict__ gout,   // (B, 128)
                     float* __restrict__ gy)     // (N, 128)
{
  __shared__ float xs  [CMAX * DFEAT];   // x tile          (TDM destination)
  __shared__ float zs  [CMAX * DFEAT];   // z state
  __shared__ float auxs[CMAX * DFEAT];   // log_aux / log_tran (reused, LOG domain)
  __shared__ float red [NWAVES][DFEAT];  // cross-wave combine buffer
  __shared__ float mu1s[CMAX];           // per-row log_mu1
  __shared__ float z1s [CMAX];           // per-row z1
  __shared__ float prk[KITER], pa1[KITER], pa2[KITER], pa3[KITER];
  __shared__ int   seg_bounds[2];

  const int b    = blockIdx.x;
  const int tid  = threadIdx.x;
  const int wv   = tid >> 5;
  const int lane = tid & 31;

  // segment range via binary search on sorted batch ids
  if (tid == 0) {
    int lo = 0, hi = N_NODES;
    while (lo < hi) { int mid = (lo + hi) >> 1; if (gbatch[mid] < b) lo = mid + 1; else hi = mid; }
    seg_bounds[0] = lo;
    int lo2 = lo; hi = N_NODES;
    while (lo2 < hi) { int mid = (lo2 + hi) >> 1; if (gbatch[mid] < b + 1) lo2 = mid + 1; else hi = mid; }
    seg_bounds[1] = lo2;
  }
  if (tid < KITER) {   // softplus'd params once, in LDS
    prk[tid] = softplus_f(grho[tid]);
    pa1[tid] = softplus_f(ga1[tid]);
    pa2[tid] = softplus_f(ga2[tid]);
    pa3[tid] = softplus_f(ga3[tid]);
  }
  __syncthreads();

  const int start = seg_bounds[0];
  const int C     = seg_bounds[1] - seg_bounds[0];   // rows in this segment (>=1)

  // ---- stage x tile (C x 128 f32) into LDS via the Tensor Data Mover ----
#if __has_builtin(__builtin_amdgcn_tensor_load_to_lds)
  if (wv == 0) {
    const unsigned long long gaddr =
        (unsigned long long)(const void*)(gx + (unsigned long long)start * DFEAT);
    const int      sC    = __builtin_amdgcn_readfirstlane((int)C);
    const unsigned ga_lo = (unsigned)__builtin_amdgcn_readfirstlane((int)(gaddr & 0xffffffffull));
    const unsigned ga_hi = (unsigned)__builtin_amdgcn_readfirstlane((int)(gaddr >> 32));
    const unsigned ldsa  = (unsigned)__builtin_amdgcn_readfirstlane(
        (int)(unsigned)(unsigned long long)(const void*)&xs[0]);
    // D# group 0: count=1 | lds_addr | global_addr[56:0] | type=2
    v4u g0 = { 1u, ldsa, ga_lo, (ga_hi & 0x01ffffffu) | (2u << 30) };
    // D# group 1: data_size=4B, tensor 128 x C, tile 128 x C, dim0_stride=128
    v8i g1 = { 0x00020000,
               (int)((DFEAT & 0xffff) << 16),   // tensor_dim0[15:0] @ bits 63:48
               (int)((sC & 0xffff) << 16),      // tensor_dim1[15:0] @ bits 95:80
               (int)(DFEAT << 16),              // tile_dim0 @ bits 127:112
               (int)(sC & 0xffff),              // tile_dim1 @ bits 143:128
               DFEAT,                           // tensor_dim0_stride[31:0]
               0, 0 };
    v4i g2 = { 0, 0, 0, 0 };
    v4i g3 = { 0, 0, 0, 0 };
#if __clang_major__ >= 23
    v8i gz = { 0, 0, 0, 0, 0, 0, 0, 0 };
    __builtin_amdgcn_tensor_load_to_lds(g0, g1, g2, g3, gz, 0);
#else
    __builtin_amdgcn_tensor_load_to_lds(g0, g1, g2, g3, 0);
#endif
  }
#endif

  // init state while the TDM transfer is in flight
  const float log_u1 = -logf((float)C);
  const float log_u2 = -4.852030263919617f;          // -log(128)
  const float la0    = log_u1 + log_u2;              // init log_aux
  for (int i = tid; i < C * DFEAT; i += NTHREADS) { zs[i] = 0.0f; auxs[i] = la0; }
  for (int r = tid; r < C; r += NTHREADS)          { mu1s[r] = log_u1; z1s[r] = 0.0f; }

#if __has_builtin(__builtin_amdgcn_tensor_load_to_lds)
  __builtin_amdgcn_s_wait_tensorcnt(0);
#else
  for (int i = tid; i < C * DFEAT; i += NTHREADS)
    xs[i] = gx[(unsigned long long)start * DFEAT + i];
#endif
  __syncthreads();

  // per-wave replicated (segment, d) state: 4 column-chunks per lane
  float mu2[4], z2l[4];
#pragma unroll
  for (int c = 0; c < 4; ++c) { mu2[c] = log_u2; z2l[c] = 0.0f; }

  const float NEG_INF = -__builtin_inff();

  // ---- iterations 0 .. K-2 (full step) ----
  for (int k = 0; k < KITER - 1; ++k) {
    const float rk  = prk[k], a1k = pa1[k], a2k = pa2[k], a3k = pa3[k];
    const float inv_rk   = 1.0f / rk;
    const float inv_a1rk = 1.0f / (a1k + rk);
    const float inv_rka2 = 1.0f / (rk + a2k);
    const float inv_rka3 = 1.0f / (rk + a3k);

    // Pass 1: t1, row-lse, log_tran (stored over log_aux), per-column t2 max
    float m2[4] = { NEG_INF, NEG_INF, NEG_INF, NEG_INF };
    for (int r = wv; r < C; r += NWAVES) {
      const int base = r * DFEAT + lane;
      const float mu1r = mu1s[r];
      float t1c[4], zc[4];
      float mx = NEG_INF;
#pragma unroll
      for (int c = 0; c < 4; ++c) {
        const int idx = base + c * 32;
        const float z = zs[idx]; zc[c] = z;
        const float t1 = auxs[idx] + (xs[idx] - z) * inv_rk;
        t1c[c] = t1; mx = fmaxf(mx, t1);
      }
      mx = wave_reduce_max(mx);
      float s = 0.0f;
#pragma unroll
      for (int c = 0; c < 4; ++c) s += expf(t1c[c] - mx);
      s = wave_reduce_sum(s);
      const float lse = logf(s) + mx;
#pragma unroll
      for (int c = 0; c < 4; ++c) {
        const int idx = base + c * 32;
        const float lt = mu1r + t1c[c] - lse;   // log_tran (uses OLD mu1)
        auxs[idx] = lt;
        const float t2 = (zc[c] + rk * lt) * inv_a1rk;
        m2[c] = fmaxf(m2[c], t2);
      }
    }
    // segment max (cross-wave)
#pragma unroll
    for (int c = 0; c < 4; ++c) red[wv][c * 32 + lane] = m2[c];
    __syncthreads();
    float Mv[4];
#pragma unroll
    for (int c = 0; c < 4; ++c) {
      float m = red[0][c * 32 + lane];
#pragma unroll
      for (int w = 1; w < NWAVES; ++w) m = fmaxf(m, red[w][c * 32 + lane]);
      Mv[c] = m;
    }
    __syncthreads();

    // Pass 2: segment sum of exp(t2 - max)
    float Sp[4] = { 0.f, 0.f, 0.f, 0.f };
    for (int r = wv; r < C; r += NWAVES) {
      const int base = r * DFEAT + lane;
#pragma unroll
      for (int c = 0; c < 4; ++c) {
        const int idx = base + c * 32;
        const float t2 = (zs[idx] + rk * auxs[idx]) * inv_a1rk;
        Sp[c] += expf(t2 - Mv[c]);
      }
    }
#pragma unroll
    for (int c = 0; c < 4; ++c) red[wv][c * 32 + lane] = Sp[c];
    __syncthreads();
    float seglse[4];
#pragma unroll
    for (int c = 0; c < 4; ++c) {
      float s = 0.0f;
#pragma unroll
      for (int w = 0; w < NWAVES; ++w) s += red[w][c * 32 + lane];
      seglse[c] = logf(s) + Mv[c];
    }
    __syncthreads();

    // Pass 3: log_aux (stored in LOG domain), tran, z / z1 / mu1 updates,
    // segment sum of aux
    float Ap[4] = { 0.f, 0.f, 0.f, 0.f };
    for (int r = wv; r < C; r += NWAVES) {
      const int base = r * DFEAT + lane;
      float trow = 0.0f;
#pragma unroll
      for (int c = 0; c < 4; ++c) {
        const int idx = base + c * 32;
        const float lt   = auxs[idx];                       // log_tran
        const float z    = zs[idx];
        const float t2   = (z + rk * lt) * inv_a1rk;
        const float laux = t2 - seglse[c] + mu2[c];         // log_aux (OLD mu2)
        const float aux  = expf(laux);
        const float tr   = expf(lt);
        Ap[c] += aux;
        trow  += tr;
        zs[idx]   = z + a1k * (tr - aux);
        auxs[idx] = laux;                                   // keep LOG domain
      }
      trow = wave_reduce_sum(trow);
      const float mu1o = mu1s[r];
      const float z1o  = z1s[r];
      const float mu1n = (rk * mu1o + a2k * log_u1 - z1o) * inv_rka2;
      const float z1n  = z1o + rk * (expf(mu1n) - trow);
      if (lane == 0) { mu1s[r] = mu1n; z1s[r] = z1n; }
    }
#pragma unroll
    for (int c = 0; c < 4; ++c) red[wv][c * 32 + lane] = Ap[c];
    __syncthreads();
#pragma unroll
    for (int c = 0; c < 4; ++c) {
      float s = 0.0f;
#pragma unroll
      for (int w = 0; w < NWAVES; ++w) s += red[w][c * 32 + lane];
      const float mu2n = (rk * mu2[c] + a3k * log_u2 - z2l[c]) * inv_rka3;
      z2l[c] += rk * (expf(mu2n) - s);
      mu2[c] = mu2n;
    }
    __syncthreads();
  }

  // ---- final step (k = K-1): only log_tran + outputs are needed ----
  {
    const float rk     = prk[KITER - 1];
    const float inv_rk = 1.0f / rk;
    float Osum[4] = { 0.f, 0.f, 0.f, 0.f };
    for (int r = wv; r < C; r += NWAVES) {
      const int base = r * DFEAT + lane;
      const float mu1r = mu1s[r];
      float t1c[4];
      float mx = NEG_INF;
#pragma unroll
      for (int c = 0; c < 4; ++c) {
        const int idx = base + c * 32;
        const float t1 = auxs[idx] + (xs[idx] - zs[idx]) * inv_rk;
        t1c[c] = t1; mx = fmaxf(mx, t1);
      }
      mx = wave_reduce_max(mx);
      float s = 0.0f;
#pragma unroll
      for (int c = 0; c < 4; ++c) s += expf(t1c[c] - mx);
      s = wave_reduce_sum(s);
      const float lse = logf(s) + mx;
#pragma unroll
      for (int c = 0; c < 4; ++c) {
        const int idx = base + c * 32;
        const float y = expf(mu1r + t1c[c] - lse) + EPS_Y;
        gy[(unsigned long long)(start + r) * DFEAT + c * 32 + lane] = y;
        Osum[c] += xs[idx] * y;
      }
    }
    // cross-wave column-sum combine of the 8x128 partials via f32 WMMA:
    // D = A(partials 16x4) * B(ones 4x16): row m of D = sum over K of partials.
    // Only the documented A and C/D layouts are relied upon (B = all-ones).
#pragma unroll
    for (int c = 0; c < 4; ++c) red[wv][c * 32 + lane] = Osum[c];
    __syncthreads();
#if __has_builtin(__builtin_amdgcn_wmma_f32_16x16x4_f32)
    {
      const int m16 = lane & 15;       // A-matrix row M held by this lane
      const int kh  = lane >> 4;       // K-half select per A layout
      v2f bones; bones.x = 1.0f; bones.y = 1.0f;
      v8f acc = { 0.f, 0.f, 0.f, 0.f, 0.f, 0.f, 0.f, 0.f };
#pragma unroll
      for (int s2 = 0; s2 < 2; ++s2) {  // K = 8 waves -> two K=4 steps
        v2f av;
        av.x = red[4 * s2 + 2 * kh + 0][16 * wv + m16];  // A vgpr0: K = 0/2
        av.y = red[4 * s2 + 2 * kh + 1][16 * wv + m16];  // A vgpr1: K = 1/3
        acc = __builtin_amdgcn_wmma_f32_16x16x4_f32(false, av, false, bones,
                                                    (short)0, acc, false, false);
      }
      // D layout: lane 0 holds D[0..7][0] in acc[0..7]; lane 16 holds D[8..15][0]
      if (lane == 0 || lane == 16) {
        const int moff = (lane == 16) ? 8 : 0;
#pragma unroll
        for (int v = 0; v < 8; ++v)
          gout[b * DFEAT + 16 * wv + moff + v] = (float)DFEAT * acc[v];
      }
    }
#else
    if (tid < DFEAT) {
      float s = 0.0f;
#pragma unroll
      for (int w = 0; w < NWAVES; ++w) s += red[w][tid];
      gout[b * DFEAT + tid] = (float)DFEAT * s;
    }
#endif
  }
}

extern "C" void kernel_launch(void* const* d_in, const int* in_sizes, int n_in,
                              void* d_out, int out_size, void* d_ws, size_t ws_size,
                              hipStream_t stream) {
  (void)in_sizes; (void)n_in; (void)out_size; (void)d_ws; (void)ws_size;
  const float* x     = (const float*)d_in[0];
  const float* rho   = (const float*)d_in[1];
  const float* a1    = (const float*)d_in[2];
  const float* a2    = (const float*)d_in[3];
  const float* a3    = (const float*)d_in[4];
  const int*   batch = (const int*)d_in[5];
  float* out = (float*)d_out;              // (B,128) first
  float* y   = out + BSEG * DFEAT;         // then (N,128)
  uot_pool_kernel<<<dim3(BSEG), dim3(NTHREADS), 0, stream>>>(
      x, rho, a1, a2, a3, batch, out, y);
}